// Dense_54047868452885
// MI455X (gfx1250) — compile-verified
//
#include <hip/hip_runtime.h>

// Problem constants (from reference): B=8192, IN=1024, OUT=1024, all fp32.
#define B_DIM   8192
#define IN_DIM  1024
#define OUT_DIM 1024

// Block tile: 64 (M) x 128 (N); 8 waves, each computing a 32x32 register tile
// (2x2 grid of 16x16 f32 WMMA accumulators). K staged through LDS in chunks of 32.
#define BM 64
#define BN 128
#define KT 32
#define LDK (KT + 4)   // row pitch 36 floats: 16B-aligned float4 rows, conflict-free banks

typedef float v2f __attribute__((ext_vector_type(2)));
typedef float v4f __attribute__((ext_vector_type(4)));
typedef float v8f __attribute__((ext_vector_type(8)));

__global__ void zero_kernel(float* __restrict__ p, int n) {
    int i = blockIdx.x * blockDim.x + threadIdx.x;
    if (i < n) p[i] = 0.0f;
}

__launch_bounds__(256)
__global__ void dense_wmma_f32(const float* __restrict__ X,
                               const float* __restrict__ W,
                               const float* __restrict__ bvec,
                               const float* __restrict__ Aux,
                               const float* __restrict__ state,
                               float* __restrict__ out,
                               float* __restrict__ colsum_g) {
    __shared__ __align__(16) float As[BM][LDK];   // [m][k]
    __shared__ __align__(16) float Bs[BN][LDK];   // [n][k] == W[n][k] (no transpose)
    __shared__ float csum[BN];

    const int tid  = threadIdx.x;
    const int wave = tid >> 5;     // 8 waves (wave32)
    const int lane = tid & 31;

    const int m0 = blockIdx.x * BM;
    const int n0 = blockIdx.y * BN;

    // Wave grid: 2 (M) x 4 (N); each wave owns a 32x32 output tile.
    const int wm = wave & 1;
    const int wn = wave >> 1;

    const int ml    = lane & 15;         // M (A frag) / N (B,C frag) sub-index
    const int khalf = (lane >> 4) * 2;   // lanes 16-31 hold K+2,K+3 (ISA 7.12.2)

    v8f acc00 = {}, acc01 = {}, acc10 = {}, acc11 = {};

    if (tid < BN) csum[tid] = 0.0f;

    // ---- Precompute staged-copy addresses (all float4, all 16B aligned) ----
    // A tile: 64x32 f32 = 512 float4, 2 per thread.
    const float* xsrc[2]; float* adst[2];
    #pragma unroll
    for (int i = 0; i < 2; ++i) {
        int idx = tid + i * 256;
        int row = idx >> 3, c4 = idx & 7;
        xsrc[i] = X + (size_t)(m0 + row) * IN_DIM + c4 * 4;
        adst[i] = &As[row][c4 * 4];
    }
    // B tile: 128x32 f32 = 1024 float4, 4 per thread.
    const float* wsrc[4]; float* bdst[4];
    #pragma unroll
    for (int i = 0; i < 4; ++i) {
        int idx = tid + i * 256;
        int n = idx >> 3, k4 = idx & 7;
        wsrc[i] = W + (size_t)(n0 + n) * IN_DIM + k4 * 4;
        bdst[i] = &Bs[n][k4 * 4];
    }

    // ---- Software pipeline: prefetch stage 0 into registers ----
    v4f av[2], bv[4];
    #pragma unroll
    for (int i = 0; i < 2; ++i) av[i] = *(const v4f*)xsrc[i];
    #pragma unroll
    for (int i = 0; i < 4; ++i) bv[i] = *(const v4f*)wsrc[i];

    const float* Ar0 = &As[wm * 32 + ml][0];
    const float* Ar1 = &As[wm * 32 + 16 + ml][0];
    const float* Br0 = &Bs[wn * 32 + ml][0];
    const float* Br1 = &Bs[wn * 32 + 16 + ml][0];

    for (int k0 = 0; k0 < IN_DIM; k0 += KT) {
        // Commit staged registers to LDS (waits only on the pending global loads).
        #pragma unroll
        for (int i = 0; i < 2; ++i) *(v4f*)adst[i] = av[i];
        #pragma unroll
        for (int i = 0; i < 4; ++i) *(v4f*)bdst[i] = bv[i];
        __syncthreads();

        // Issue next stage's global loads; they retire during the WMMA block below.
        if (k0 + KT < IN_DIM) {
            #pragma unroll
            for (int i = 0; i < 2; ++i) av[i] = *(const v4f*)(xsrc[i] + k0 + KT);
            #pragma unroll
            for (int i = 0; i < 4; ++i) bv[i] = *(const v4f*)(wsrc[i] + k0 + KT);
        }

        // 8 K-steps x 4 WMMA: 2 A frags + 2 B frags feed a 2x2 accumulator tile.
        #pragma unroll
        for (int kk = 0; kk < KT; kk += 4) {
            const int kb = kk + khalf;
            v2f a0 = { Ar0[kb], Ar0[kb + 1] };
            v2f a1 = { Ar1[kb], Ar1[kb + 1] };
            v2f b0 = { Br0[kb], Br0[kb + 1] };
            v2f b1 = { Br1[kb], Br1[kb + 1] };
            acc00 = __builtin_amdgcn_wmma_f32_16x16x4_f32(
                false, a0, false, b0, (short)0, acc00, false, false);
            acc01 = __builtin_amdgcn_wmma_f32_16x16x4_f32(
                false, a0, false, b1, (short)0, acc01, false, false);
            acc10 = __builtin_amdgcn_wmma_f32_16x16x4_f32(
                false, a1, false, b0, (short)0, acc10, false, false);
            acc11 = __builtin_amdgcn_wmma_f32_16x16x4_f32(
                false, a1, false, b1, (short)0, acc11, false, false);
        }
        __syncthreads();
    }

    // ---- Epilogue: fuse bias + Aux0*state, store, column partial sums ----
    // C/D layout (ISA 7.12.2): VGPR r -> M = r + 8*(lane>=16), N = lane&15.
    const int mh  = (lane >> 4) * 8;
    const int ng0 = n0 + wn * 32 + ml;
    const int ng1 = ng0 + 16;
    const float e0 = bvec[ng0] + Aux[2 * ng0] * state[ng0];
    const float e1 = bvec[ng1] + Aux[2 * ng1] * state[ng1];
    const int mgb0 = m0 + wm * 32 + mh;
    const int mgb1 = mgb0 + 16;

    float s0 = 0.0f, s1 = 0.0f;
    #pragma unroll
    for (int r = 0; r < 8; ++r) {
        out[(size_t)(mgb0 + r) * OUT_DIM + ng0] = acc00[r] + e0;
        out[(size_t)(mgb0 + r) * OUT_DIM + ng1] = acc01[r] + e1;
        out[(size_t)(mgb1 + r) * OUT_DIM + ng0] = acc10[r] + e0;
        out[(size_t)(mgb1 + r) * OUT_DIM + ng1] = acc11[r] + e1;
        s0 += acc00[r] + acc10[r];
        s1 += acc01[r] + acc11[r];
    }
    atomicAdd(&csum[wn * 32 + ml], s0);
    atomicAdd(&csum[wn * 32 + 16 + ml], s1);
    __syncthreads();
    if (tid < BN) atomicAdd(&colsum_g[n0 + tid], csum[tid]);
}

// new_state[o] = Aux[o,1]*state[o] + colsum[o]/B + b[o]   (mean of constant bias == bias)
__global__ void finalize_state(const float* __restrict__ colsum,
                               const float* __restrict__ bvec,
                               const float* __restrict__ Aux,
                               const float* __restrict__ state,
                               float* __restrict__ new_state) {
    int o = blockIdx.x * blockDim.x + threadIdx.x;
    if (o < OUT_DIM) {
        new_state[o] = Aux[2 * o + 1] * state[o]
                     + colsum[o] * (1.0f / (float)B_DIM)
                     + bvec[o];
    }
}

extern "C" void kernel_launch(void* const* d_in, const int* in_sizes, int n_in,
                              void* d_out, int out_size, void* d_ws, size_t ws_size,
                              hipStream_t stream) {
    const float* X     = (const float*)d_in[0];  // [B, IN]
    const float* W     = (const float*)d_in[1];  // [OUT, IN]
    const float* bvec  = (const float*)d_in[2];  // [OUT]
    const float* Aux   = (const float*)d_in[3];  // [OUT, 2]
    const float* state = (const float*)d_in[4];  // [OUT]
    // d_in[5] = depth (unused by the math)

    float* out       = (float*)d_out;                       // [B, OUT]
    float* new_state = out + (size_t)B_DIM * OUT_DIM;       // [OUT]
    float* colsum    = (float*)d_ws;                        // [OUT] scratch (4 KB)

    zero_kernel<<<(OUT_DIM + 255) / 256, 256, 0, stream>>>(colsum, OUT_DIM);

    dim3 grid(B_DIM / BM, OUT_DIM / BN);   // 128 x 8 blocks
    dense_wmma_f32<<<grid, 256, 0, stream>>>(X, W, bvec, Aux, state, out, colsum);

    finalize_state<<<(OUT_DIM + 255) / 256, 256, 0, stream>>>(
        colsum, bvec, Aux, state, new_state);
}